// FieldLoss_25340307046655
// MI455X (gfx1250) — compile-verified
//
#include <hip/hip_runtime.h>
#include <hip/hip_bf16.h>
#include <math.h>

typedef float v2f __attribute__((ext_vector_type(2)));
typedef float v8f __attribute__((ext_vector_type(8)));

#define PRE 64
#define BATCH 1024
#define NNB 256
#define MASSC 60.0f

// D = A(16x4 f32) * B(4x16 f32) + 0, wave32 WMMA.
// 8-arg form: (neg_a, A, neg_b, B, c_mod, C, reuse_a, reuse_b)
__device__ __forceinline__ v8f wmma_k4(v2f a, v2f b) {
  v8f c = {};
  return __builtin_amdgcn_wmma_f32_16x16x4_f32(false, a, false, b, (short)0, c,
                                               false, false);
}

__global__ __launch_bounds__(256) void field_block_kernel(
    const float* __restrict__ output, const float* __restrict__ target,
    const float* __restrict__ nbr, float* __restrict__ ws) {
  __shared__ float px[PRE], py[PRE];
  // B-operand tables packed by lane-half so phase 2 is branchless:
  //  BxyT[0*NNB+n] = (ca, sa, -sa, ca)   (rows K0,K1 of Bx | By)
  //  BxyT[1*NNB+n] = (bx, 0,  by,  0 )   (rows K2,K3 of Bx | By)
  //  mnT[n]        = (-nx, -ny)          (row K2 of Bdx | Bdy)
  __shared__ float4 BxyT[2 * NNB];
  __shared__ float2 mnT[NNB];
  __shared__ float exh[2 * PRE], eyh[2 * PRE];
  __shared__ float red[256];

  const int b = blockIdx.x;
  const int tid = threadIdx.x;

  // ---- phase 0: MSE partial + stage trajectory (px,py) in LDS ----
  float msePart = 0.f;
  if (tid < 2 * PRE) {
    int t = tid >> 1, c = tid & 1;
    int idx = (t * BATCH + b) * 2 + c;
    float o = output[idx];
    float d = o - target[idx];
    msePart = d * d;
    if (c == 0) px[t] = o; else py[t] = o;
  }

  // ---- phase 1: neighbour coefficient tables (one n per thread) ----
  {
    const float* p = nbr + (size_t)(b * NNB + tid) * 5;
    float nx = p[0], ny = p[1], ang = p[4];
    float s, c;
    __sincosf(ang, &s, &c);
    float bx = -(c * nx + s * ny);  // K2 coeff of x_ = ca*dx+sa*dy
    float by = s * nx - c * ny;     // K2 coeff of y_ = -sa*dx+ca*dy
    BxyT[tid]       = make_float4(c, s, -s, c);
    BxyT[NNB + tid] = make_float4(bx, 0.f, by, 0.f);
    mnT[tid]        = make_float2(-nx, -ny);
  }

  // ---- MSE block reduce (deterministic tree) ----
  red[tid] = msePart;
  __syncthreads();
  for (int s = 128; s > 0; s >>= 1) {
    if (tid < s) red[tid] += red[tid + s];
    __syncthreads();
  }
  if (tid == 0) ws[b] = red[0];

  // ---- phase 2: WMMA tiles ----
  // wave w: t-tile = w&3, n-tiles [(w>>2)*8, +8)
  const int wave = tid >> 5;
  const int lane = tid & 31;
  const int laneLo = lane & 15;
  const bool hiHalf = (lane >= 16);
  const int t0 = (wave & 3) * 16;
  const int ntBase = (wave >> 2) * 8;
  const int bOff = hiHalf ? NNB : 0;

  // A (16x4, K=(px,py,1,0)): VGPR0 = K0|K2, VGPR1 = K1|K3 per lane halves
  v2f A;
  A.x = hiHalf ? 1.0f : px[t0 + laneLo];
  A.y = hiHalf ? 0.0f : py[t0 + laneLo];

  float exA[8], eyA[8];
#pragma unroll
  for (int i = 0; i < 8; ++i) { exA[i] = 0.f; eyA[i] = 0.f; }

#pragma unroll 2
  for (int nt = 0; nt < 8; ++nt) {
    const int nc = (ntBase + nt) * 16 + laneLo;
    // Branchless B operands: one b128 + one b64 LDS load, value-selects only.
    float4 bxy = BxyT[bOff + nc];
    float2 mn = mnT[nc];
    v2f Bx, By, Bdx, Bdy;
    Bx.x = bxy.x;  Bx.y = bxy.y;
    By.x = bxy.z;  By.y = bxy.w;
    Bdx.x = hiHalf ? mn.x : 1.0f;  Bdx.y = 0.0f;
    Bdy.x = hiHalf ? mn.y : 0.0f;  Bdy.y = hiHalf ? 0.0f : 1.0f;

    v8f X  = wmma_k4(A, Bx);
    v8f Y  = wmma_k4(A, By);
    v8f DX = wmma_k4(A, Bdx);
    v8f DY = wmma_k4(A, Bdy);
#pragma unroll
    for (int i = 0; i < 8; ++i) {
      float x = X[i], y = Y[i], dx = DX[i], dy = DY[i];
      float x2 = x * x, y2 = y * y;
      float den = (x2 * x2) * x2 + (y2 * y2) * y2 + 6.0f;
      float e = MASSC * __builtin_amdgcn_rcpf(den);
      // cos(atan2(-dy,-dx)) = -dx/r ; sin = -dy/r
      float r2 = dx * dx + dy * dy;
      float rin = __builtin_amdgcn_rsqf(fmaxf(r2, 1e-30f));
      exA[i] -= e * dx * rin;
      eyA[i] -= e * dy * rin;
    }
  }

  // reduce across the 16 n-lanes of each half (D layout: N = lane&15)
#pragma unroll
  for (int i = 0; i < 8; ++i) {
    float ex = exA[i], ey = eyA[i];
#pragma unroll
    for (int m = 1; m <= 8; m <<= 1) {
      ex += __shfl_xor(ex, m, 32);
      ey += __shfl_xor(ey, m, 32);
    }
    if (laneLo == 0) {
      int t = t0 + i + (hiHalf ? 8 : 0);  // D: M = i + 8*(lane>=16)
      int half = wave >> 2;
      exh[half * PRE + t] = ex;
      eyh[half * PRE + t] = ey;
    }
  }
  __syncthreads();

  // ---- phase 3: per-t norm + block reduce ----
  float en = 0.f;
  if (tid < PRE) {
    float ex = exh[tid] + exh[PRE + tid];
    float ey = eyh[tid] + eyh[PRE + tid];
    en = sqrtf(ex * ex + ey * ey);
  }
  red[tid] = en;
  __syncthreads();
  for (int s = 128; s > 0; s >>= 1) {
    if (tid < s) red[tid] += red[tid + s];
    __syncthreads();
  }
  if (tid == 0) ws[BATCH + b] = red[0];
}

__global__ __launch_bounds__(256) void finalize_kernel(
    const float* __restrict__ ws, float* __restrict__ out) {
  __shared__ float red[512];
  const int tid = threadIdx.x;
  float mse = 0.f, fe = 0.f;
  for (int i = tid; i < BATCH; i += 256) {
    mse += ws[i];
    fe += ws[BATCH + i];
  }
  red[tid] = mse;
  red[256 + tid] = fe;
  __syncthreads();
  for (int s = 128; s > 0; s >>= 1) {
    if (tid < s) {
      red[tid] += red[tid + s];
      red[256 + tid] += red[256 + tid + s];
    }
    __syncthreads();
  }
  if (tid == 0) {
    out[0] = red[0] / (float)(PRE * BATCH * 2) + red[256] / (float)(PRE * BATCH);
  }
}

extern "C" void kernel_launch(void* const* d_in, const int* in_sizes, int n_in,
                              void* d_out, int out_size, void* d_ws,
                              size_t ws_size, hipStream_t stream) {
  (void)in_sizes; (void)n_in; (void)out_size; (void)ws_size;
  const float* output = (const float*)d_in[0];   // (64,1024,2)
  const float* target = (const float*)d_in[1];   // (64,1024,2)
  const float* nbr    = (const float*)d_in[2];   // (1024,256,5)
  float* out = (float*)d_out;
  float* ws  = (float*)d_ws;  // [0,1024): mse partials; [1024,2048): fe partials

  field_block_kernel<<<BATCH, 256, 0, stream>>>(output, target, nbr, ws);
  finalize_kernel<<<1, 256, 0, stream>>>(ws, out);
}